// GCN_7172595384593
// MI455X (gfx1250) — compile-verified
//
#include <hip/hip_runtime.h>

typedef __attribute__((ext_vector_type(2))) float v2f;
typedef __attribute__((ext_vector_type(8))) float v8f;

// ------------------------------------------------------------------
// Kernel 1: zero the h[N*HID] accumulator in workspace (required every
// call: SpMM accumulates with atomics and the harness replays the graph).
// ------------------------------------------------------------------
__global__ void gcn_zero_kernel(float* __restrict__ h, long long n) {
    long long i = (long long)blockIdx.x * blockDim.x + threadIdx.x;
    long long stride = (long long)gridDim.x * blockDim.x;
    for (; i < n; i += stride) h[i] = 0.0f;
}

// ------------------------------------------------------------------
// Kernel 2: edge-parallel SpMM (gather + scale + segment-sum).
// One wave per edge; edge scalars fetched via readfirstlane so they
// lower to SMEM loads; features gathered as float2 (global_load_b64)
// and scattered with global_atomic_add_f32. W0 (40MB) and h (40MB)
// both stay resident in the 192MB L2.
// ------------------------------------------------------------------
__global__ void gcn_spmm_kernel(const int* __restrict__ rows,
                                const int* __restrict__ cols,
                                const float* __restrict__ vals,
                                const float* __restrict__ W0,
                                float* __restrict__ h,
                                int E, int HID) {
    const int wavesPerBlock = blockDim.x >> 5;
    const int wave = threadIdx.x >> 5;
    const int lane = threadIdx.x & 31;
    int e = blockIdx.x * wavesPerBlock + wave;
    if (e >= E) return;                       // wave-uniform exit
    e = __builtin_amdgcn_readfirstlane(e);    // force SGPR -> scalar loads

    const int   r = rows[e];
    const int   c = cols[e];
    const float v = vals[e];

    const float* __restrict__ src = W0 + (long long)c * HID;
    float* __restrict__ dst       = h  + (long long)r * HID;

    // pairs of features: lane handles f = 2*lane + 64*i (HID=200 is even)
    for (int f = lane * 2; f + 1 < HID; f += 64) {
        const float2 w = *(const float2*)&src[f];
        atomicAdd(&dst[f],     v * w.x);
        atomicAdd(&dst[f + 1], v * w.y);
    }
    // odd-HID tail (not taken for HID=200)
    if ((HID & 1) && lane == 0) {
        atomicAdd(&dst[HID - 1], v * src[HID - 1]);
    }
}

// ------------------------------------------------------------------
// Kernel 3: out = relu(h) @ W2 via V_WMMA_F32_16X16X4_F32.
// W2 is staged once per block into LDS, zero-padded C->16 columns and
// transposed (sB[n*HID + k]) so inner-loop B operands are unconditional
// ds_load_b64. Full tiles (the only kind for N%16==0) take a fast path
// whose A operand is just fmaxf(load, 0) -> no guard ALU in the hot
// loop; the generic tail path keeps a branch-free 0/1 row neutralizer.
// EXEC stays all-1s through every WMMA (wave-uniform branches only).
//
// VGPR layouts (CDNA5 ISA 7.12.2):
//   A 16x4 f32 : lane l holds M=(l&15); v0 = K=(l>>4)*2, v1 = K=+1
//   B 4x16 f32 : lane l holds N=(l&15); v0 = K=(l>>4)*2, v1 = K=+1
//   C/D 16x16  : lane l holds N=(l&15); vi = M=(l>>4)*8 + i
// ------------------------------------------------------------------
#define MAX_HID 256   // HID=200 in this workload; sB = 16*256*4 = 16 KB LDS

__global__ void gcn_gemm_wmma_kernel(const float* __restrict__ h,
                                     const float* __restrict__ W2,
                                     float* __restrict__ out,
                                     int N, int HID, int C) {
    __shared__ float sB[16 * MAX_HID];

    // cooperative stage: sB[n*HID + k] = (n < C) ? W2[k*C + n] : 0
    for (int i = threadIdx.x; i < 16 * HID; i += blockDim.x) {
        const int n = i / HID;
        const int k = i - n * HID;
        sB[i] = (n < C) ? W2[k * C + n] : 0.0f;
    }
    __syncthreads();

    const int wave = threadIdx.x >> 5;
    const int lane = threadIdx.x & 31;
    const int tile = blockIdx.x * (blockDim.x >> 5) + wave;
    const int m0   = tile * 16;
    if (m0 >= N) return;                  // wave-uniform exit after barrier

    const int laneM = lane & 15;          // M for A, N for B/C/D
    const int kOff  = (lane >> 4) << 1;   // 0 or 2: which K-pair this half holds
    const float* __restrict__ bRow = sB + laneM * HID;

    v8f acc = {};

    if (m0 + 16 <= N) {
        // -------- fast path: full 16-row tile, no guards in the loop ----
        const float* __restrict__ aRow = h + (long long)(m0 + laneM) * HID;
        for (int kk = 0; kk < HID; kk += 4) {
            const float2 av = *(const float2*)&aRow[kk + kOff]; // global_load_b64
            const float2 bv = *(const float2*)&bRow[kk + kOff]; // ds_load_b64
            v2f a, b;
            a.x = fmaxf(av.x, 0.0f);      // fused ReLU
            a.y = fmaxf(av.y, 0.0f);
            b.x = bv.x;
            b.y = bv.y;
            acc = __builtin_amdgcn_wmma_f32_16x16x4_f32(
                false, a, false, b, (short)0, acc, false, false);
        }
    } else {
        // -------- tail path: clamp row, branch-free 0/1 neutralizer -----
        const int   rowA     = m0 + laneM;
        const float rowScale = (rowA < N) ? 1.0f : 0.0f;
        const int   rowClamp = (rowA < N) ? rowA : (N - 1);
        const float* __restrict__ aRow = h + (long long)rowClamp * HID;
        for (int kk = 0; kk < HID; kk += 4) {
            const float2 av = *(const float2*)&aRow[kk + kOff];
            const float2 bv = *(const float2*)&bRow[kk + kOff];
            v2f a, b;
            a.x = fmaxf(av.x, 0.0f) * rowScale;
            a.y = fmaxf(av.y, 0.0f) * rowScale;
            b.x = bv.x;
            b.y = bv.y;
            acc = __builtin_amdgcn_wmma_f32_16x16x4_f32(
                false, a, false, b, (short)0, acc, false, false);
        }
    }

    // Store D: lane l owns column (l&15), rows m0 + (l>>4)*8 + i
    const int col   = laneM;
    const int mBase = (lane >> 4) << 3;
    if (col < C) {
        #pragma unroll
        for (int i = 0; i < 8; ++i) {
            const int row = m0 + mBase + i;
            if (row < N) out[(long long)row * C + col] = acc[i];
        }
    }
}

// ------------------------------------------------------------------
// Launch: inputs per setup_inputs() order:
//   d_in[0]=x[N,1] d_in[1]=rows[E] d_in[2]=cols[E] d_in[3]=vals[E]
//   d_in[4]=W0[N,HID] d_in[5]=W2[HID,C]
// ------------------------------------------------------------------
extern "C" void kernel_launch(void* const* d_in, const int* in_sizes, int n_in,
                              void* d_out, int out_size, void* d_ws, size_t ws_size,
                              hipStream_t stream) {
    const int*   rows = (const int*)  d_in[1];
    const int*   cols = (const int*)  d_in[2];
    const float* vals = (const float*)d_in[3];
    const float* W0   = (const float*)d_in[4];
    const float* W2   = (const float*)d_in[5];
    float*       out  = (float*)d_out;

    const int N   = in_sizes[0];            // x is [N,1]
    const int E   = in_sizes[1];
    const int HID = (int)(in_sizes[4] / (long long)N);
    const int C   = (int)(in_sizes[5] / (long long)HID);

    float* h = (float*)d_ws;                // needs N*HID*4 = 40 MB of scratch
    const long long hElems = (long long)N * HID;

    {   // zero accumulator
        const int threads = 256;
        long long b = (hElems + threads - 1) / threads;
        if (b > 262144) b = 262144;         // grid-stride covers the rest
        gcn_zero_kernel<<<(int)b, threads, 0, stream>>>(h, hElems);
    }
    {   // SpMM: one wave per edge, 8 waves per 256-thread block
        const int threads = 256;
        const int wpb = threads / 32;
        const int blocks = (E + wpb - 1) / wpb;
        gcn_spmm_kernel<<<blocks, threads, 0, stream>>>(rows, cols, vals, W0, h, E, HID);
    }
    {   // WMMA GEMM: one wave per 16-row tile
        const int threads = 256;
        const int wpb = threads / 32;
        const int tiles = (N + 15) / 16;
        const int blocks = (tiles + wpb - 1) / wpb;
        gcn_gemm_wmma_kernel<<<blocks, threads, 0, stream>>>(h, W2, out, N, HID, C);
    }
}